// UM_loss_21234318311857
// MI455X (gfx1250) — compile-verified
//
#include <hip/hip_runtime.h>
#include <hip/hip_bf16.h>
#include <math.h>

// ---------------- problem constants ----------------
#define BATCH 32
#define TLEN  750
#define DIM   2048
#define NCLS  20
#define MNODE 750
#define NSAMP 64
#define MPAD  768          // padded M stride for cross matrix (multiple of 128)

#define ALPHA      0.0005f
#define BETA       0.1f
#define MARGIN     100.0f
#define GCN_WEIGHT 0.1f
#define EPSV       1e-8f

// ---------------- workspace layout (floats) ----------------
#define WS_CLS   0
#define WS_BE    1
#define WS_UM    2
#define WS_GCN   3
#define WS_SSACT 8                      // 32 floats
#define WS_SSBKG 40                     // 32 floats
#define WS_ACCN  128                    // accumulator region size (zeroed each call)
#define WS_N2    128                    // BATCH*MNODE floats
#define WS_CROSS 24576                  // BATCH*NSAMP*MPAD floats

typedef __attribute__((ext_vector_type(2))) float v2f;
typedef __attribute__((ext_vector_type(8))) float v8f;

// ---------------- helpers ----------------
__device__ __forceinline__ float warpSum(float v) {
#pragma unroll
  for (int o = 16; o > 0; o >>= 1) v += __shfl_xor(v, o, 32);
  return v;
}

__device__ __forceinline__ float blockSum256(float v, float* sbuf) {
  int lane = threadIdx.x & 31, w = threadIdx.x >> 5;
  v = warpSum(v);
  if (lane == 0) sbuf[w] = v;
  __syncthreads();
  v = (threadIdx.x < 8) ? sbuf[threadIdx.x] : 0.0f;
  if (w == 0) {
#pragma unroll
    for (int o = 4; o > 0; o >>= 1) v += __shfl_xor(v, o, 32);
  }
  __syncthreads();
  return v;  // valid on thread 0
}

// ---------------- kernel: zero accumulators ----------------
__global__ void k_init(float* ws) {
  if (threadIdx.x < WS_ACCN) ws[threadIdx.x] = 0.0f;
}

// ---------------- kernel: BCE losses (tiny) ----------------
__global__ void k_bce(const float* __restrict__ score_act,
                      const float* __restrict__ score_bkg,
                      const float* __restrict__ label,
                      float* __restrict__ ws) {
  __shared__ float rowsum[BATCH];
  __shared__ float sbuf[8];
  int tid = threadIdx.x;
  if (tid < BATCH) {
    float s = 0.0f;
    for (int j = 0; j < NCLS; ++j) s += label[tid * NCLS + j];
    rowsum[tid] = s;
  }
  __syncthreads();
  float cls = 0.0f, be = 0.0f;
  const float tb = 1.0f / (float)NCLS;
  for (int i = tid; i < BATCH * NCLS; i += blockDim.x) {
    int b = i / NCLS;
    float t = label[i] / rowsum[b];
    float p = score_act[i];
    cls += t * __logf(p) + (1.0f - t) * __logf(1.0f - p);
    float pb = score_bkg[i];
    be += tb * __logf(pb) + (1.0f - tb) * __logf(1.0f - pb);
  }
  float clsS = blockSum256(cls, sbuf);
  float beS  = blockSum256(be, sbuf);
  if (tid == 0) {
    ws[WS_CLS] = -clsS / (float)(BATCH * NCLS);
    ws[WS_BE]  = -beS  / (float)(BATCH * NCLS);
  }
}

// ---------------- kernel: loss_um column reductions ----------------
// grid: (DIM/256, BATCH, 2); block 256. z=0 -> feat_act, z=1 -> feat_bkg.
__global__ void k_um_cols(const float* __restrict__ feat_act,
                          const float* __restrict__ feat_bkg,
                          float* __restrict__ ws) {
  __shared__ float sbuf[8];
  int z = blockIdx.z;
  int b = blockIdx.y;
  int d = blockIdx.x * blockDim.x + threadIdx.x;
  const float* feat = (z == 0) ? feat_act : feat_bkg;
  const float* base = feat + (size_t)b * TLEN * DIM + d;
  float s = 0.0f;
  for (int t = 0; t < TLEN; ++t) s += base[(size_t)t * DIM];
  float mean = s * (1.0f / (float)TLEN);
  float ss = blockSum256(mean * mean, sbuf);
  if (threadIdx.x == 0)
    atomicAdd(&ws[(z == 0 ? WS_SSACT : WS_SSBKG) + b], ss);
}

// ---------------- kernel: loss_um finalize (1 warp) ----------------
__global__ void k_um_fin(float* __restrict__ ws) {
  int b = threadIdx.x;  // 32 lanes
  float la = fmaxf(MARGIN - sqrtf(ws[WS_SSACT + b]), 0.0f);
  float lb = sqrtf(ws[WS_SSBKG + b]);
  float v = (la + lb) * (la + lb);
  v = warpSum(v);
  if (b == 0) ws[WS_UM] = v / (float)BATCH;
}

// ---------------- kernel: batched GEMM cross = s_nodes * nodes^T  (+ fused n2) ----------------
// fp32 WMMA 16x16x4. grid: (MPAD/128, BATCH); block 256 (8 waves).
// Block covers 64 s-rows x 128 m-cols. Each wave owns 4 16x16 output tiles
// (one s-tile x 4 m-tiles) -> A fragment reused 4x per K-step.
// B-tile loader threads stream every nodes[b,m,:] row once per batch, so
// n2[b,m] = ||nodes[b,m]||^2 is accumulated for free in the loaders.
#define LDST 36  // LDS row stride in floats: 16B-aligned, conflict-free column reads

__global__ void k_cross_wmma(const float* __restrict__ nodes,
                             const int* __restrict__ sample_idx,
                             float* __restrict__ ws) {
  __shared__ float As[64 * LDST];     //  64 s-rows x 32 k
  __shared__ float Bs[128 * LDST];    // 128 m-rows x 32 k
  __shared__ int   sidx[NSAMP];

  const int b  = blockIdx.y;
  const int m0 = blockIdx.x * 128;
  const int tid  = threadIdx.x;
  const int lane = tid & 31;
  const int wave = tid >> 5;

  if (tid < NSAMP) sidx[tid] = sample_idx[b * NSAMP + tid];
  __syncthreads();

  // A loader role: row 0..63, quad 0..3, 8 floats each
  const int arow = tid >> 2;
  const int aq   = tid & 3;
  // B loader role: row 0..127, half 0..1, 16 floats each
  const int brow = tid >> 1;
  const int bq   = tid & 1;

  const int sBase  = (wave >> 1) * 16;      // s tile base for this wave (0..48)
  const int mGroup = (wave & 1) * 64;       // local m group base (0 or 64)
  const int hk     = (lane >> 4) << 1;      // 0 or 2 (K half select, frag layout)
  const int rA     = (sBase + (lane & 15)) * LDST;
  const int rBn    = (mGroup + (lane & 15)) * LDST;

  const int   mrow  = m0 + brow;            // global m row this thread loads
  const bool  bValid = (mrow < MNODE);
  const float* aSrc = nodes + ((size_t)b * MNODE + sidx[arow]) * DIM + aq * 8;
  const float* bSrc = nodes + ((size_t)b * MNODE + (bValid ? mrow : MNODE - 1)) * DIM + bq * 16;

  v8f acc0 = {}, acc1 = {}, acc2 = {}, acc3 = {};
  float sq = 0.0f;   // fused n2 accumulator (this thread's 16 B columns)

  for (int kc = 0; kc < DIM; kc += 32) {
    float4 a0 = *(const float4*)(aSrc + kc);
    float4 a1 = *(const float4*)(aSrc + kc + 4);
    float4 b0 = *(const float4*)(bSrc + kc);
    float4 b1 = *(const float4*)(bSrc + kc + 4);
    float4 b2 = *(const float4*)(bSrc + kc + 8);
    float4 b3 = *(const float4*)(bSrc + kc + 12);
    if (!bValid) {
      b0 = make_float4(0, 0, 0, 0); b1 = b0; b2 = b0; b3 = b0;
    }
    sq += b0.x * b0.x + b0.y * b0.y + b0.z * b0.z + b0.w * b0.w
        + b1.x * b1.x + b1.y * b1.y + b1.z * b1.z + b1.w * b1.w
        + b2.x * b2.x + b2.y * b2.y + b2.z * b2.z + b2.w * b2.w
        + b3.x * b3.x + b3.y * b3.y + b3.z * b3.z + b3.w * b3.w;
    *(float4*)&As[arow * LDST + aq * 8]      = a0;
    *(float4*)&As[arow * LDST + aq * 8 + 4]  = a1;
    *(float4*)&Bs[brow * LDST + bq * 16]     = b0;
    *(float4*)&Bs[brow * LDST + bq * 16 + 4] = b1;
    *(float4*)&Bs[brow * LDST + bq * 16 + 8] = b2;
    *(float4*)&Bs[brow * LDST + bq * 16 + 12] = b3;
    __syncthreads();

#pragma unroll
    for (int k4 = 0; k4 < 32; k4 += 4) {
      v2f a = { As[rA + k4 + hk], As[rA + k4 + hk + 1] };
      v2f f0 = { Bs[rBn + k4 + hk], Bs[rBn + k4 + hk + 1] };
      acc0 = __builtin_amdgcn_wmma_f32_16x16x4_f32(false, a, false, f0,
                                                   (short)0, acc0, false, false);
      v2f f1 = { Bs[rBn + 16 * LDST + k4 + hk], Bs[rBn + 16 * LDST + k4 + hk + 1] };
      acc1 = __builtin_amdgcn_wmma_f32_16x16x4_f32(false, a, false, f1,
                                                   (short)0, acc1, false, false);
      v2f f2 = { Bs[rBn + 32 * LDST + k4 + hk], Bs[rBn + 32 * LDST + k4 + hk + 1] };
      acc2 = __builtin_amdgcn_wmma_f32_16x16x4_f32(false, a, false, f2,
                                                   (short)0, acc2, false, false);
      v2f f3 = { Bs[rBn + 48 * LDST + k4 + hk], Bs[rBn + 48 * LDST + k4 + hk + 1] };
      acc3 = __builtin_amdgcn_wmma_f32_16x16x4_f32(false, a, false, f3,
                                                   (short)0, acc3, false, false);
    }
    __syncthreads();
  }

  // fused n2 write: reduce across the bq pair (adjacent lanes), one store per row
  sq += __shfl_xor(sq, 1, 32);
  if (bq == 0 && bValid) ws[WS_N2 + b * MNODE + mrow] = sq;

  // store D tiles: vgpr v -> s row = sBase + v + (lane<16 ? 0 : 8); col = lane&15
  float* cross = ws + WS_CROSS;
  const int sOff = sBase + ((lane >> 4) << 3);
  const int col0 = m0 + mGroup + (lane & 15);
#pragma unroll
  for (int v = 0; v < 8; ++v) {
    size_t base = ((size_t)(b * NSAMP + sOff + v)) * MPAD;
    cross[base + col0]      = acc0[v];
    cross[base + col0 + 16] = acc1[v];
    cross[base + col0 + 32] = acc2[v];
    cross[base + col0 + 48] = acc3[v];
  }
}

// ---------------- kernel: GCN scan (one wave per (b,s)) ----------------
// grid: (NSAMP, BATCH); block 32.
__global__ void k_gcn_scan(const int* __restrict__ labels,
                           const int* __restrict__ sample_idx,
                           float* __restrict__ ws) {
  const int b = blockIdx.y, s = blockIdx.x, lane = threadIdx.x;
  const float* n2 = ws + WS_N2 + b * MNODE;
  const float* cr = ws + WS_CROSS + ((size_t)(b * NSAMP + s)) * MPAD;
  const int sid  = sample_idx[b * NSAMP + s];
  const int slab = labels[b * MNODE + sid];
  const float s2 = n2[sid];

  float bestP = -1e30f; int idxP = 0x7fffffff;
  float bestN =  1e30f; int idxN = 0x7fffffff;
  for (int m = lane; m < MNODE; m += 32) {
    int lab = labels[b * MNODE + m];
    bool valid = lab >= 0;
    float d2 = fmaxf(s2 + n2[m] - 2.0f * cr[m], 0.0f);
    bool same = valid && (lab == slab);
    bool opp  = valid && (lab != slab);
    if (same && (d2 > bestP || (d2 == bestP && m < idxP))) { bestP = d2; idxP = m; }
    if (opp  && (d2 < bestN || (d2 == bestN && m < idxN))) { bestN = d2; idxN = m; }
  }
#pragma unroll
  for (int o = 16; o > 0; o >>= 1) {
    float ov = __shfl_xor(bestP, o, 32); int oi = __shfl_xor(idxP, o, 32);
    if (ov > bestP || (ov == bestP && oi < idxP)) { bestP = ov; idxP = oi; }
    float nv = __shfl_xor(bestN, o, 32); int ni = __shfl_xor(idxN, o, 32);
    if (nv < bestN || (nv == bestN && ni < idxN)) { bestN = nv; idxN = ni; }
  }
  if (lane == 0) {
    float sn = sqrtf(s2);
    float denom0 = fmaxf(sn, EPSV);
    float total = 0.0f;
    if (bestP > -1e29f) {  // any same-label node
      float nb = sqrtf(n2[idxP]);
      total += cr[idxP] / (denom0 * fmaxf(nb, EPSV));
    }
    if (bestN < 1e29f) {   // any opposite-label node
      float nb = sqrtf(n2[idxN]);
      total += cr[idxN] / (denom0 * fmaxf(nb, EPSV));
    }
    atomicAdd(&ws[WS_GCN], total);
  }
}

// ---------------- kernel: combine ----------------
__global__ void k_combine(const float* __restrict__ ws, float* __restrict__ out) {
  if (threadIdx.x == 0)
    out[0] = ws[WS_CLS] + ALPHA * ws[WS_UM] + BETA * ws[WS_BE] +
             GCN_WEIGHT * ws[WS_GCN];
}

// ---------------- host launcher ----------------
extern "C" void kernel_launch(void* const* d_in, const int* in_sizes, int n_in,
                              void* d_out, int out_size, void* d_ws, size_t ws_size,
                              hipStream_t stream) {
  const float* score_act  = (const float*)d_in[0];
  const float* score_bkg  = (const float*)d_in[1];
  const float* feat_act   = (const float*)d_in[2];
  const float* feat_bkg   = (const float*)d_in[3];
  const float* label      = (const float*)d_in[4];
  // d_in[5]=gt, d_in[6]=cas : unused by the reference loss
  const float* nodes      = (const float*)d_in[7];
  const int*   nodes_lab  = (const int*)d_in[8];
  const int*   sample_idx = (const int*)d_in[9];
  float* ws  = (float*)d_ws;
  float* out = (float*)d_out;

  k_init<<<1, 256, 0, stream>>>(ws);
  k_bce<<<1, 256, 0, stream>>>(score_act, score_bkg, label, ws);
  k_um_cols<<<dim3(DIM / 256, BATCH, 2), 256, 0, stream>>>(feat_act, feat_bkg, ws);
  k_um_fin<<<1, 32, 0, stream>>>(ws);
  k_cross_wmma<<<dim3(MPAD / 128, BATCH), 256, 0, stream>>>(nodes, sample_idx, ws);
  k_gcn_scan<<<dim3(NSAMP, BATCH), 32, 0, stream>>>(nodes_lab, sample_idx, ws);
  k_combine<<<1, 32, 0, stream>>>(ws, out);
}